// S4DKernel_49950469652762
// MI455X (gfx1250) — compile-verified
//
#include <hip/hip_runtime.h>
#include <hip/hip_bf16.h>
#include <math.h>

// S4D kernel for MI455X (gfx1250), wave32 + V_WMMA_F32_16X16X4_F32.
//
// K[h,l] = 2*Re( sum_n Cc'[h,n] * exp(dtA[h,n]*l) ),  H=1024, N/2=32, L=4096.
// Factor l = 64*l1 + l2  ->  per-head 64x64x64 real GEMM (K-dim = [Re|Im] pairs),
// which cuts transcendentals 32x and runs the contraction on the WMMA pipe.

typedef float v2f __attribute__((ext_vector_type(2)));
typedef float v8f __attribute__((ext_vector_type(8)));

#define HH   1024
#define NC   32        // complex modes per head (N/2)
#define LL   4096
#define T2   64        // l2 tile width (inner index)
#define LDSP 80        // padded LDS row stride (floats): bank-conflict-free frags

__global__ __launch_bounds__(128)
void s4d_wmma_f32_kernel(const float* __restrict__ log_dt,
                         const float* __restrict__ C,
                         const float* __restrict__ log_A_real,
                         const float* __restrict__ A_imag,
                         float* __restrict__ out)
{
    __shared__ float Qs[64 * LDSP];   // B matrix: [k][l2], k<32 = Re(Q), k>=32 = -Im(Q)
    __shared__ float As[64 * LDSP];   // A^T:      [k][l1], k<32 = Re(a'), k>=32 = Im(a')
    __shared__ float s_re[NC], s_im[NC], s_cr[NC], s_ci[NC];

    const int h    = blockIdx.x;
    const int tid  = threadIdx.x;
    const int lane = tid & 31;
    const int wave = tid >> 5;

    // ---------------- stage 1: per-mode scalars (threads 0..31) ----------------
    if (tid < NC) {
        const int n = tid;
        const float dt = __expf(log_dt[h]);
        const float Ar = -__expf(log_A_real[h * NC + n]);
        const float Ai = A_imag[h * NC + n];
        const float dtAr = Ar * dt, dtAi = Ai * dt;
        // w = exp(dtA)
        float s, c;
        const float e = __expf(dtAr);
        __sincosf(dtAi, &s, &c);
        const float wr = e * c, wi = e * s;
        // Cc' = 2 * Cc * (w - 1)/A ; (w-1)/A = (w-1)*conj(A)/|A|^2  (|A|^2 >= 0.25)
        const float inv = 1.0f / (Ar * Ar + Ai * Ai);
        const float gr = ((wr - 1.0f) * Ar + wi * Ai) * inv;
        const float gi = (wi * Ar - (wr - 1.0f) * Ai) * inv;
        const float Cr = C[h * 64 + n], Ci = C[h * 64 + NC + n];
        s_cr[n] = 2.0f * (Cr * gr - Ci * gi);
        s_ci[n] = 2.0f * (Cr * gi + Ci * gr);
        s_re[n] = dtAr;
        s_im[n] = dtAi;
    }
    __syncthreads();

    // ---------------- stage 2: build B (Qs) and A^T (As) in LDS ----------------
    // 32 modes x 64 positions, 128 threads -> 16 iterations each.
    for (int i = tid; i < NC * 64; i += 128) {
        const int n = i >> 6;          // mode 0..31
        const int t = i & 63;          // l2 (for Q) / l1 (for P)
        const float re = s_re[n], im = s_im[n];
        // Q[n,l2] = exp(dtA * l2)
        {
            float s, c;
            const float e = __expf(re * (float)t);
            __sincosf(im * (float)t, &s, &c);
            Qs[n * LDSP + t]        =  e * c;   //  Re(Q)
            Qs[(n + NC) * LDSP + t] = -e * s;   // -Im(Q)
        }
        // a'[l1,n] = Cc'[n] * exp(dtA * 64 * l1)
        {
            const float ft = (float)(t << 6);
            float s, c;
            const float e = __expf(re * ft);
            __sincosf(im * ft, &s, &c);
            const float pr = e * c, pi = e * s;
            const float cr = s_cr[n], ci = s_ci[n];
            As[n * LDSP + t]        = cr * pr - ci * pi;  // Re(a')
            As[(n + NC) * LDSP + t] = cr * pi + ci * pr;  // Im(a')
        }
    }
    __syncthreads();

    // ---------------- stage 3: 64x64x64 GEMM via WMMA f32 16x16x4 --------------
    // wave -> 16-row l1 strip; 4 column tiles (l2). ISA 7.12.2 fragment layouts:
    //   A 16x4 : lane%16 = M ; VGPR0 -> K = 2*(lane/16), VGPR1 -> K+1
    //   B 4x16 : lane%16 = N ; VGPR0 -> K = 2*(lane/16), VGPR1 -> K+1
    //   C/D    : VGPR v -> row v + 8*(lane/16), col lane%16
    v8f acc0 = {}, acc1 = {}, acc2 = {}, acc3 = {};
    const int m     = (wave << 4) + (lane & 15);  // A-frag row (l1)
    const int cn    = lane & 15;                  // col within tile (l2 % 16)
    const int khalf = (lane >> 4) << 1;           // 0 or 2

    #pragma unroll
    for (int ks = 0; ks < 16; ++ks) {
        const int k0 = (ks << 2) + khalf;
        v2f a, b0, b1, b2, b3;
        a.x  = As[k0 * LDSP + m];
        a.y  = As[(k0 + 1) * LDSP + m];
        b0.x = Qs[k0 * LDSP + cn];        b0.y = Qs[(k0 + 1) * LDSP + cn];
        b1.x = Qs[k0 * LDSP + 16 + cn];   b1.y = Qs[(k0 + 1) * LDSP + 16 + cn];
        b2.x = Qs[k0 * LDSP + 32 + cn];   b2.y = Qs[(k0 + 1) * LDSP + 32 + cn];
        b3.x = Qs[k0 * LDSP + 48 + cn];   b3.y = Qs[(k0 + 1) * LDSP + 48 + cn];
        acc0 = __builtin_amdgcn_wmma_f32_16x16x4_f32(false, a, false, b0, (short)0, acc0, false, false);
        acc1 = __builtin_amdgcn_wmma_f32_16x16x4_f32(false, a, false, b1, (short)0, acc1, false, false);
        acc2 = __builtin_amdgcn_wmma_f32_16x16x4_f32(false, a, false, b2, (short)0, acc2, false, false);
        acc3 = __builtin_amdgcn_wmma_f32_16x16x4_f32(false, a, false, b3, (short)0, acc3, false, false);
    }

    // ---------------- store: out[h*4096 + l1*64 + l2] --------------------------
    const int rbase = (wave << 4) + ((lane >> 4) << 3);      // D row base
    float* op = out + (size_t)h * LL + rbase * T2 + cn;
    #pragma unroll
    for (int v = 0; v < 8; ++v) {
        float* row = op + v * T2;
        row[0]  = acc0[v];
        row[16] = acc1[v];
        row[32] = acc2[v];
        row[48] = acc3[v];
    }
}

extern "C" void kernel_launch(void* const* d_in, const int* in_sizes, int n_in,
                              void* d_out, int out_size, void* d_ws, size_t ws_size,
                              hipStream_t stream) {
    const float* log_dt     = (const float*)d_in[0];
    const float* C          = (const float*)d_in[1];
    const float* log_A_real = (const float*)d_in[2];
    const float* A_imag     = (const float*)d_in[3];
    // d_in[4] is L (scalar int) — fixed at compile time (4096).
    float* out = (float*)d_out;
    s4d_wmma_f32_kernel<<<dim3(HH), dim3(128), 0, stream>>>(
        log_dt, C, log_A_real, A_imag, out);
}